// KBpA_54451595379052
// MI455X (gfx1250) — compile-verified
//
#include <hip/hip_runtime.h>
#include <hip/hip_bf16.h>

// Problem constants (match reference)
#define HH 128
#define WW 128
#define NPIX (HH * WW)         // 16384 pixels
#define NB 16                  // batches
#define KG 1024                // grid centers (32x32, every 4 px)
#define KS 21                  // Gaussian kernel size
#define KHALF 10

typedef __attribute__((ext_vector_type(2))) float v2f;
typedef __attribute__((ext_vector_type(8))) float v8f;

// ---------------------------------------------------------------------------
// Stage 1: deformation field.
// def[b, p, ch] = sum over nearby grid centers g of betas[b,g,ch] * K[r-gr+10, c-gc+10]
// Grid spacing 4, kernel half-width 10 -> at most 6x6 contributing centers.
// ---------------------------------------------------------------------------
__global__ void deform_kernel(const float* __restrict__ betas,
                              const float* __restrict__ kern,
                              float* __restrict__ def) {
    int idx = blockIdx.x * blockDim.x + threadIdx.x;   // 0 .. NB*NPIX-1
    if (idx >= NB * NPIX) return;
    int b = idx >> 14;          // / NPIX
    int p = idx & (NPIX - 1);
    int r = p >> 7;
    int c = p & (WW - 1);

    // grid row indices gi with |r - 4*gi| <= 10  -> gi in [ceil((r-10)/4), floor((r+10)/4)]
    int gi_lo = (r > KHALF - 3) ? ((r - (KHALF - 3)) >> 2) : 0;   // ceil((r-10)/4) clamped
    int gi_hi = (r + KHALF) >> 2; if (gi_hi > 31) gi_hi = 31;
    int gj_lo = (c > KHALF - 3) ? ((c - (KHALF - 3)) >> 2) : 0;
    int gj_hi = (c + KHALF) >> 2; if (gj_hi > 31) gj_hi = 31;

    float acc0 = 0.0f, acc1 = 0.0f;
    for (int gi = gi_lo; gi <= gi_hi; ++gi) {
        int kr = r - (gi << 2) + KHALF;                // 0..20
        const float* krow = kern + kr * KS;
        for (int gj = gj_lo; gj <= gj_hi; ++gj) {
            int kc = c - (gj << 2) + KHALF;            // 0..20
            float w = krow[kc];
            int g = (b * KG + (gi << 5) + gj) * 2;
            acc0 = fmaf(w, betas[g + 0], acc0);
            acc1 = fmaf(w, betas[g + 1], acc1);
        }
    }
    float* o = def + (size_t)idx * 2;
    o[0] = acc0;
    o[1] = acc1;
}

// ---------------------------------------------------------------------------
// Stage 2: Gaussian kernel sum via V_WMMA_F32_16X16X4_F32.
// log2-domain exponent(n,m) = [dx,dy,dx^2+dy^2,1] .
//     L2E * [cx/25, cy/25, -1/50, -(cx^2+cy^2)/50]
// log2(e) is baked into the B operand so the WMMA result feeds v_exp_f32
// directly (no per-element base conversion multiply).
// One wave handles 16 pixels of one batch, loops over 64 tiles of 16 centers.
// ---------------------------------------------------------------------------
__global__ void __launch_bounds__(256)
gauss_sum_kernel(const float* __restrict__ def,
                 const float* __restrict__ alphas,
                 float* __restrict__ out) {
    const int lane = threadIdx.x & 31;
    const int wave = threadIdx.x >> 5;                   // 8 waves / block
    const int tile = blockIdx.x * 8 + wave;              // 0 .. NB*(NPIX/16)-1
    const int b    = tile >> 10;                         // NPIX/16 = 1024 tiles/batch
    const int n0   = (tile & 1023) << 4;                 // first pixel of tile

    // Per-lane pixel for the A operand (lanes 0-15 and 16-31 mirror the same rows)
    const int p  = n0 + (lane & 15);
    const int r  = p >> 7;
    const int c  = p & (WW - 1);
    const float* dptr = def + ((size_t)b * NPIX + p) * 2;
    const float dx = (float)r - dptr[0];
    const float dy = (float)c - dptr[1];
    const float sq = fmaf(dx, dx, dy * dy);

    // 32-bit A-matrix 16x4: lanes 0-15 -> K0,K1 in v0,v1 ; lanes 16-31 -> K2,K3
    const bool hi = lane >= 16;
    v2f a;
    a.x = hi ? sq   : dx;
    a.y = hi ? 1.0f : dy;

    float acc[8] = {0.f, 0.f, 0.f, 0.f, 0.f, 0.f, 0.f, 0.f};
    const float L2E   = 1.4426950408889634f;
    const float C_XY  = 0.04f * L2E;                     // (2/50) * log2(e)
    const float C_SQ  = -0.02f * L2E;                    // (-1/50) * log2(e)

    for (int mt = 0; mt < KG / 16; ++mt) {
        const int m  = (mt << 4) + (lane & 15);          // center index for this column
        const float cx = (float)((m >> 5) << 2) + 2.0f;  // row-major 32x32 grid, step 4, +2
        const float cy = (float)((m & 31) << 2) + 2.0f;
        const float sqc = fmaf(cx, cx, cy * cy);

        // 32-bit B-matrix 4x16: lanes 0-15 -> K0,K1 ; lanes 16-31 -> K2,K3
        // log2(e) pre-folded into every coefficient.
        v2f bb;
        bb.x = hi ? C_SQ       : cx * C_XY;
        bb.y = hi ? sqc * C_SQ : cy * C_XY;

        v8f d = {};
        d = __builtin_amdgcn_wmma_f32_16x16x4_f32(
                /*neg_a=*/false, a, /*neg_b=*/false, bb,
                /*c_mod=*/(short)0, d, /*reuse_a=*/false, /*reuse_b=*/false);

        const float alpha = alphas[m];                   // column weight, uniform per lane
#pragma unroll
        for (int k = 0; k < 8; ++k)
            acc[k] = fmaf(alpha, __builtin_amdgcn_exp2f(d[k]), acc[k]);
    }

    // Reduce across the 16 columns (lanes 0-15 and 16-31 independently).
#pragma unroll
    for (int k = 0; k < 8; ++k) {
        float v = acc[k];
        v += __shfl_xor(v, 1, 32);
        v += __shfl_xor(v, 2, 32);
        v += __shfl_xor(v, 4, 32);
        v += __shfl_xor(v, 8, 32);
        acc[k] = v;
    }

    // VGPR k of lane-group holds rows k (lanes 0-15) and k+8 (lanes 16-31).
    if ((lane & 15) == 0) {
        const int rowbase = (lane >> 4) << 3;            // 0 or 8
        float4* o = (float4*)(out + (size_t)b * NPIX + n0 + rowbase);
        o[0] = make_float4(acc[0], acc[1], acc[2], acc[3]);
        o[1] = make_float4(acc[4], acc[5], acc[6], acc[7]);
    }
}

// ---------------------------------------------------------------------------
extern "C" void kernel_launch(void* const* d_in, const int* in_sizes, int n_in,
                              void* d_out, int out_size, void* d_ws, size_t ws_size,
                              hipStream_t stream) {
    const float* betas  = (const float*)d_in[0];   // (16, 1024, 2)
    const float* alphas = (const float*)d_in[1];   // (1024, 1)
    const float* kern   = (const float*)d_in[2];   // (21, 21)
    float*       out    = (float*)d_out;           // (16, 16384, 1)
    float*       def    = (float*)d_ws;            // (16, 16384, 2) scratch = 2 MB

    // Stage 1: deformation field
    {
        const int threads = 256;
        const int total = NB * NPIX;
        deform_kernel<<<(total + threads - 1) / threads, threads, 0, stream>>>(
            betas, kern, def);
    }

    // Stage 2: WMMA-tiled Gaussian kernel sum (log2-domain exponents)
    {
        const int tiles = NB * (NPIX / 16);        // 16384 waves
        const int threads = 256;                   // 8 waves / block
        gauss_sum_kernel<<<tiles / 8, threads, 0, stream>>>(def, alphas, out);
    }
}